// MPNN_87454124081721
// MI455X (gfx1250) — compile-verified
//
#include <hip/hip_runtime.h>

typedef __attribute__((ext_vector_type(2))) float v2f;
typedef __attribute__((ext_vector_type(8))) float v8f;

#if defined(__has_builtin) && __has_builtin(__builtin_amdgcn_sched_barrier)
#define SCHED_FENCE() __builtin_amdgcn_sched_barrier(0)
#else
#define SCHED_FENCE()
#endif

#define N_NODES 50000
#define N_EDGES 500000
#define D_NODE 64
#define D_EDGE 32
#define MSG_DIM 160
#define HID_DIM 224
#define ALPHA 0.01f

// Padded LDS row strides: stride % 64 == 36 -> conflict-free row access
#define LDA_MSG 164
#define LDA_HID 228

#define EDGE_WAVES 10                    // MSG_DIM / 16 output tiles
#define EDGE_THREADS (EDGE_WAVES * 32)
#define EDGE_TILES_PER_BLOCK 10
#define EDGE_KSTEPS (MSG_DIM / 4)        // 40
#define EDGE_KCHUNK 20                   // 2 chunks: batch LDS prefetch per chunk

#define NODE_WAVES 14                    // HID_DIM / 16 output tiles
#define NODE_THREADS (NODE_WAVES * 32)
#define NODE_TILES_PER_BLOCK 5
#define NODE_KSTEPS (HID_DIM / 4)        // 56
#define NODE_KCHUNK 28                   // 2 chunks

__device__ __forceinline__ float leaky(float x) {
    return fmaxf(x, 0.0f) + ALPHA * fminf(x, 0.0f);
}

__global__ void mpnn_zero_kernel(float* __restrict__ p, int n4) {
    int i = blockIdx.x * blockDim.x + threadIdx.x;
    const int stride = gridDim.x * blockDim.x;
    const float4 z = make_float4(0.f, 0.f, 0.f, 0.f);
    for (; i < n4; i += stride) ((float4*)p)[i] = z;
}

// ---------------------------------------------------------------------------
// Edge phase: m = leaky(concat(h_n[src], h_n[dst], h_e) @ W_msg^T + b_msg)
//             m_sum[dst] += m        (fp32 atomics, L2-resident target)
// One block = 16 edges x 160 outputs; wave w owns output cols [16w, 16w+16).
// ---------------------------------------------------------------------------
__global__ __launch_bounds__(EDGE_THREADS)
void mpnn_edge_msg_kernel(const float* __restrict__ h_n,
                          const float* __restrict__ h_e,
                          const int*   __restrict__ src,
                          const int*   __restrict__ dst,
                          const float* __restrict__ W_msg,
                          const float* __restrict__ b_msg,
                          float*       __restrict__ m_sum)
{
    __shared__ float As[16 * LDA_MSG];

    const int tid   = threadIdx.x;
    const int wave  = tid >> 5;
    const int lane  = tid & 31;
    const int col   = lane & 15;            // N (and A-row M) index within tile
    const int khalf = (lane >> 4) << 1;     // 0 or 2: K sub-offset for this half-wave
    const int n0    = wave << 4;            // first output column of this wave

    // B[k][n] = W_msg[n][k]; preload this wave's K=160 x N=16 slice (80 VGPRs),
    // reused across all edge tiles handled by this block.
    const float* wrow = W_msg + (size_t)(n0 + col) * MSG_DIM + khalf;
    v2f breg[EDGE_KSTEPS];
#pragma unroll
    for (int t = 0; t < EDGE_KSTEPS; ++t) {
        const float2 w = *(const float2*)(wrow + 4 * t);
        breg[t].x = w.x;
        breg[t].y = w.y;
    }

    const float bias = b_msg[n0 + col];

    const int tile0 = blockIdx.x * EDGE_TILES_PER_BLOCK;
    for (int it = 0; it < EDGE_TILES_PER_BLOCK; ++it) {
        const int base = (tile0 + it) << 4;              // first edge of tile

        // ---- stage A = [h_n[src] | h_n[dst] | h_e] as 16 x 160 into LDS ----
        for (int idx = tid; idx < 16 * 40; idx += EDGE_THREADS) {
            const int row = idx / 40;                    // edge within tile
            const int c4  = idx % 40;                    // float4 column
            const int e   = base + row;
            float4 v;
            if (c4 < 16)
                v = ((const float4*)(h_n + (size_t)src[e] * D_NODE))[c4];
            else if (c4 < 32)
                v = ((const float4*)(h_n + (size_t)dst[e] * D_NODE))[c4 - 16];
            else
                v = ((const float4*)(h_e + (size_t)e * D_EDGE))[c4 - 32];
            *(float4*)&As[row * LDA_MSG + c4 * 4] = v;
        }
        __syncthreads();

        // ---- 16x16 fp32 output tile: 40 chained V_WMMA_F32_16X16X4_F32.
        //      A fragments are prefetched per chunk; sched_barrier pins the
        //      batch of ds_loads ahead of the WMMA run so the waitcnt pass
        //      can overlap LDS latency with the WMMA chain. ----
        v8f acc;
#pragma unroll
        for (int i = 0; i < 8; ++i) acc[i] = bias;

        const float* arow = &As[col * LDA_MSG + khalf];
#pragma unroll
        for (int c = 0; c < EDGE_KSTEPS / EDGE_KCHUNK; ++c) {
            v2f af[EDGE_KCHUNK];
#pragma unroll
            for (int j = 0; j < EDGE_KCHUNK; ++j) {
                const int t = c * EDGE_KCHUNK + j;
                af[j].x = arow[4 * t];
                af[j].y = arow[4 * t + 1];
            }
            SCHED_FENCE();   // keep the load batch above the WMMA run
#pragma unroll
            for (int j = 0; j < EDGE_KCHUNK; ++j) {
                acc = __builtin_amdgcn_wmma_f32_16x16x4_f32(
                    false, af[j], false, breg[c * EDGE_KCHUNK + j],
                    (short)0, acc, false, false);
            }
            SCHED_FENCE();
        }

        // ---- LeakyReLU + scatter-add into m_sum[dst] ----
#pragma unroll
        for (int v = 0; v < 8; ++v) {
            const int m = v + ((lane >> 4) << 3);        // D row for this VGPR
            const int e = base + m;
            atomicAdd(&m_sum[(size_t)dst[e] * MSG_DIM + n0 + col], leaky(acc[v]));
        }
        __syncthreads();
    }
}

// ---------------------------------------------------------------------------
// Node phase: h_out = leaky(concat(m_sum, h_n) @ W_hid^T + b_hid)
// One block = 16 nodes x 224 outputs; wave w owns cols [16w, 16w+16).
// ---------------------------------------------------------------------------
__global__ __launch_bounds__(NODE_THREADS)
void mpnn_node_update_kernel(const float* __restrict__ h_n,
                             const float* __restrict__ m_sum,
                             const float* __restrict__ W_hid,
                             const float* __restrict__ b_hid,
                             float*       __restrict__ h_out)
{
    __shared__ float As[16 * LDA_HID];

    const int tid   = threadIdx.x;
    const int wave  = tid >> 5;
    const int lane  = tid & 31;
    const int col   = lane & 15;
    const int khalf = (lane >> 4) << 1;
    const int n0    = wave << 4;

    const float bias = b_hid[n0 + col];
    const float* wrow = W_hid + (size_t)(n0 + col) * HID_DIM + khalf;

    const int tile0 = blockIdx.x * NODE_TILES_PER_BLOCK;
    for (int it = 0; it < NODE_TILES_PER_BLOCK; ++it) {
        const int base = (tile0 + it) << 4;              // first node of tile

        // ---- stage A = [m_sum | h_n] as 16 x 224 into LDS ----
        for (int idx = tid; idx < 16 * 56; idx += NODE_THREADS) {
            const int row = idx / 56;
            const int c4  = idx % 56;
            const int nd  = base + row;
            float4 v;
            if (c4 < 40)
                v = ((const float4*)(m_sum + (size_t)nd * MSG_DIM))[c4];
            else
                v = ((const float4*)(h_n + (size_t)nd * D_NODE))[c4 - 40];
            *(float4*)&As[row * LDA_HID + c4 * 4] = v;
        }
        __syncthreads();

        // ---- 16x16 fp32 output tile: 56 chained f32 WMMAs,
        //      A (LDS) and B (global, L2-hot) prefetched per chunk and
        //      pinned above the WMMA run via sched_barrier ----
        v8f acc;
#pragma unroll
        for (int i = 0; i < 8; ++i) acc[i] = bias;

        const float* arow = &As[col * LDA_HID + khalf];
#pragma unroll
        for (int c = 0; c < NODE_KSTEPS / NODE_KCHUNK; ++c) {
            v2f af[NODE_KCHUNK];
            v2f bf[NODE_KCHUNK];
#pragma unroll
            for (int j = 0; j < NODE_KCHUNK; ++j) {
                const int t = c * NODE_KCHUNK + j;
                af[j].x = arow[4 * t];
                af[j].y = arow[4 * t + 1];
                const float2 w = *(const float2*)(wrow + 4 * t);
                bf[j].x = w.x;
                bf[j].y = w.y;
            }
            SCHED_FENCE();   // keep the load batch above the WMMA run
#pragma unroll
            for (int j = 0; j < NODE_KCHUNK; ++j) {
                acc = __builtin_amdgcn_wmma_f32_16x16x4_f32(
                    false, af[j], false, bf[j], (short)0, acc, false, false);
            }
            SCHED_FENCE();
        }

        // ---- LeakyReLU + store ----
#pragma unroll
        for (int v = 0; v < 8; ++v) {
            const int m = v + ((lane >> 4) << 3);
            h_out[(size_t)(base + m) * HID_DIM + n0 + col] = leaky(acc[v]);
        }
        __syncthreads();
    }
}

extern "C" void kernel_launch(void* const* d_in, const int* in_sizes, int n_in,
                              void* d_out, int out_size, void* d_ws, size_t ws_size,
                              hipStream_t stream) {
    (void)in_sizes; (void)n_in; (void)out_size; (void)ws_size;

    const float* h_n   = (const float*)d_in[0];
    const float* h_e   = (const float*)d_in[1];
    const int*   src   = (const int*)d_in[2];
    const int*   dst   = (const int*)d_in[3];
    const float* W_msg = (const float*)d_in[4];
    const float* b_msg = (const float*)d_in[5];
    const float* W_hid = (const float*)d_in[6];
    const float* b_hid = (const float*)d_in[7];

    float* h_out = (float*)d_out;
    float* m_sum = (float*)d_ws;                 // N_NODES * MSG_DIM fp32 = 32 MB

    // 1) zero the segment-sum accumulator
    mpnn_zero_kernel<<<2048, 256, 0, stream>>>(m_sum, N_NODES * MSG_DIM / 4);

    // 2) edge messages + atomic scatter-sum   (31250 edge tiles / 10 per block)
    mpnn_edge_msg_kernel<<<N_EDGES / 16 / EDGE_TILES_PER_BLOCK, EDGE_THREADS, 0, stream>>>(
        h_n, h_e, src, dst, W_msg, b_msg, m_sum);

    // 3) node update                          (3125 node tiles / 5 per block)
    mpnn_node_update_kernel<<<N_NODES / 16 / NODE_TILES_PER_BLOCK, NODE_THREADS, 0, stream>>>(
        h_n, m_sum, W_hid, b_hid, h_out);
}